// DIYloss_1709396984424
// MI455X (gfx1250) — compile-verified
//
#include <hip/hip_runtime.h>
#include <hip/hip_bf16.h>

// CDNA5 (gfx1250) wave32 WMMA-based reduction for the DIY pairwise ranking loss.
//
// Algebraic collapse of the O(L^2) pairwise sum into 5 length-L reductions:
//   pair_sum = n1*n0 + n1*S0_2 + n0*S1_2 + 2*n1*S0_1 - 2*n0*S1_1 - 2*S1_1*S0_1
// where S1_x = sum over positive labels of p^x, S0_x = Sp_x - S1_x.
//
// Single workgroup (1024 threads = 32 wave32s on one WGP), one streaming pass
// (64 KB total -> launch-latency bound on MI455X's 23.3 TB/s HBM). Both the
// intra-wave and the cross-wave reductions run on the matrix unit via
// V_WMMA_F32_16X16X4_F32 with A = partials, B = ones (exact f32 row-sums).

#define LBL 8192

typedef __attribute__((ext_vector_type(2))) float v2f;
typedef __attribute__((ext_vector_type(8))) float v8f;

// Full 32-lane sum using the matrix unit.
// A: VGPR0 = x, VGPR1 = 0  ->  A[m][0]=x[m], A[m][2]=x[m+16], others 0.
// B: all ones (layout independent).
// D[m][n] = x[m] + x[m+16]. Each lane sums its 8 accumulator VGPRs
// (M = 0..7 for lanes 0-15, M = 8..15 for lanes 16-31), then one
// half-swap shuffle merges the two halves -> total in every lane.
__device__ __forceinline__ float wave_sum_wmma(float x) {
    v2f a;
    a.x = x;
    a.y = 0.0f;
    v2f b;
    b.x = 1.0f;
    b.y = 1.0f;
    v8f c = {};
    c = __builtin_amdgcn_wmma_f32_16x16x4_f32(
        /*neg_a=*/false, a, /*neg_b=*/false, b,
        /*c_mod=*/(short)0, c, /*reuse_a=*/false, /*reuse_b=*/false);
    float t = ((c[0] + c[1]) + (c[2] + c[3])) + ((c[4] + c[5]) + (c[6] + c[7]));
    t += __shfl_xor(t, 16, 32);
    return t;
}

// Fast sigmoid: v_exp_f32 + v_rcp_f32 (~1 ulp), avoids the IEEE divide
// expansion (v_div_scale/v_div_fmas/v_div_fixup chains).
__device__ __forceinline__ float fast_sigmoid(float x) {
    return __builtin_amdgcn_rcpf(1.0f + __expf(-x));
}

__global__ __launch_bounds__(1024) void diyloss_kernel(
    const float* __restrict__ pred,
    const int* __restrict__ label,
    float* __restrict__ out) {
    const int tid = threadIdx.x;  // 0..1023
    const int lane = tid & 31;
    const int wave = tid >> 5;

    float sp = 0.0f, sp2 = 0.0f, s11 = 0.0f, s12 = 0.0f, c1 = 0.0f;

    // 8192 elements / 1024 threads = 8 per thread, as 2 x float4 / int4
    // coalesced b128 loads (block covers 16 KB per iteration).
    const float4* p4 = (const float4*)pred;
    const int4* y4 = (const int4*)label;
#pragma unroll
    for (int i = 0; i < 2; ++i) {
        const int idx = tid + i * 1024;
        const float4 x = p4[idx];
        const int4 y = y4[idx];
        const float px[4] = {x.x, x.y, x.z, x.w};
        const int yy[4] = {y.x, y.y, y.z, y.w};
#pragma unroll
        for (int j = 0; j < 4; ++j) {
            const float p = fast_sigmoid(px[j]);
            const float pp = p * p;
            const float m = (yy[j] > 0) ? 1.0f : 0.0f;
            sp += p;
            sp2 += pp;
            s11 += m * p;
            s12 += m * pp;
            c1 += m;
        }
    }

    // Stage 1: wave-level (32-lane) reductions on the matrix unit.
    sp = wave_sum_wmma(sp);
    sp2 = wave_sum_wmma(sp2);
    s11 = wave_sum_wmma(s11);
    s12 = wave_sum_wmma(s12);
    c1 = wave_sum_wmma(c1);

    // Per-wave partials -> LDS (separate arrays: conflict-free lane-stride-1).
    __shared__ float sm_sp[32], sm_sp2[32], sm_s11[32], sm_s12[32], sm_c1[32];
    if (lane == 0) {
        sm_sp[wave] = sp;
        sm_sp2[wave] = sp2;
        sm_s11[wave] = s11;
        sm_s12[wave] = s12;
        sm_c1[wave] = c1;
    }
    __syncthreads();

    // Stage 2: cross-wave combine, again on the matrix unit. Every wave does
    // the identical reduction redundantly -> no divergence, EXEC stays all-1s
    // at every WMMA (ISA requirement), no serial thread-0 loop.
    const float Sp = wave_sum_wmma(sm_sp[lane]);
    const float Sp2 = wave_sum_wmma(sm_sp2[lane]);
    const float S11 = wave_sum_wmma(sm_s11[lane]);
    const float S12 = wave_sum_wmma(sm_s12[lane]);
    const float N1 = wave_sum_wmma(sm_c1[lane]);

    if (tid == 0) {
        const float N0 = (float)LBL - N1;
        const float S01 = Sp - S11;   // sum p over zero labels
        const float S02 = Sp2 - S12;  // sum p^2 over zero labels
        const float pair_sum = N1 * N0 + N1 * S02 + N0 * S12 +
                               2.0f * N1 * S01 - 2.0f * N0 * S11 -
                               2.0f * S11 * S01;
        const float denom = (N1 * N0 > 0.0f) ? (N1 * N0) : 1.0f;
        const float zero_loss = Sp2 / (float)LBL;
        out[0] = (N1 == 0.0f) ? zero_loss : (pair_sum / denom);
    }
}

extern "C" void kernel_launch(void* const* d_in, const int* in_sizes, int n_in,
                              void* d_out, int out_size, void* d_ws, size_t ws_size,
                              hipStream_t stream) {
    (void)in_sizes;
    (void)n_in;
    (void)out_size;
    (void)d_ws;
    (void)ws_size;
    const float* pred = (const float*)d_in[0];
    const int* label = (const int*)d_in[1];
    float* out = (float*)d_out;
    diyloss_kernel<<<1, 1024, 0, stream>>>(pred, label, out);
}